// BPRsimple_60765197304485
// MI455X (gfx1250) — compile-verified
//
#include <hip/hip_runtime.h>
#include <hip/hip_bf16.h>

typedef __attribute__((ext_vector_type(2))) float v2f;
typedef __attribute__((ext_vector_type(8))) float v8f;

#define BATCH   256
#define NCOL    512
#define KPOS    64
#define NNEG    (NCOL - KPOS)   /* 448 */
#define THREADS 256

// Numerically stable softplus(d) = max(d,0) + log(1 + exp(-|d|)),
// using the hardware transcendental path (v_exp_f32 / v_log_f32).
__device__ __forceinline__ float softplus_f32(float d) {
    return fmaxf(d, 0.0f) + __logf(1.0f + __expf(-fabsf(d)));
}

__global__ __launch_bounds__(THREADS)
void bpr_row_kernel(const float* __restrict__ pred,
                    const int*   __restrict__ target,
                    float*       __restrict__ partial) {
    __shared__ float sPred[NCOL];
    __shared__ float sMask[NCOL];
    __shared__ float sRed[THREADS];

    const int t = threadIdx.x;
    const int b = blockIdx.x;
    const int rowBase = b * NCOL;

    // Stage the full row in LDS: value + validity mask (2 elems / thread).
    #pragma unroll
    for (int k = 0; k < 2; ++k) {
        const int j = t + k * THREADS;
        sPred[j] = pred[rowBase + j];
        sMask[j] = (target[rowBase + j] == -1) ? 0.0f : 1.0f;
    }
    __syncthreads();

    // Each thread owns up to 2 negative columns, kept in registers.
    const float n0  = sPred[KPOS + t];
    const float mn0 = sMask[KPOS + t];
    const bool  has2 = (t < NNEG - THREADS);               // t < 192
    const float n1  = has2 ? sPred[KPOS + THREADS + t] : 0.0f;
    const float mn1 = has2 ? sMask[KPOS + THREADS + t] : 0.0f;

    float acc = 0.0f;
    #pragma unroll 8
    for (int i = 0; i < KPOS; ++i) {
        const float pi = sPred[i];   // uniform LDS reads
        const float mi = sMask[i];
        acc += mi * (mn0 * softplus_f32(n0 - pi)
                   + mn1 * softplus_f32(n1 - pi));
    }
    sRed[t] = acc;
    __syncthreads();

    // Block reduction of the 256 partials in wave 0 via four chained
    // exact-f32 WMMAs: D = A(16x4 data) x B(4x16 ones) + C.
    // 4 chains x 64 values = all 256 partials folded into 16 row sums
    // (replicated across every column of D since B is all ones).
    if (t < 32) {
        const int lane = t;  // EXEC all-ones within wave 0
        v2f ones;
        ones[0] = 1.0f; ones[1] = 1.0f;
        v8f c = {};
        #pragma unroll
        for (int chain = 0; chain < 4; ++chain) {
            v2f a;
            a[0] = sRed[chain * 64 + lane];
            a[1] = sRed[chain * 64 + 32 + lane];
            c = __builtin_amdgcn_wmma_f32_16x16x4_f32(
                    /*neg_a=*/false, a, /*neg_b=*/false, ones,
                    /*c_mod=*/(short)0, c,
                    /*reuse_a=*/false, /*reuse_b=*/false);
        }
        // Lane L<16 holds rows M=0..7 in c[0..7]; lane L+16 holds M=8..15.
        // Columns are identical (B=ones), so total = col sum at lane0 + lane16.
        float s = c[0] + c[1] + c[2] + c[3] + c[4] + c[5] + c[6] + c[7];
        const float total = __shfl(s, 0, 32) + __shfl(s, 16, 32);
        if (lane == 0) partial[b] = total;
    }
}

// Deterministic final reduction: one wave32 sums 256 partials in a fixed
// order, scales by 1/BATCH (the mean), writes the scalar output.
__global__ __launch_bounds__(32)
void bpr_final_kernel(const float* __restrict__ partial,
                      float*       __restrict__ out) {
    const int lane = threadIdx.x;
    float s = 0.0f;
    #pragma unroll
    for (int k = 0; k < BATCH / 32; ++k)
        s += partial[lane + k * 32];
    #pragma unroll
    for (int off = 16; off >= 1; off >>= 1)
        s += __shfl_down(s, off, 32);
    if (lane == 0)
        out[0] = s * (1.0f / (float)BATCH);
}

extern "C" void kernel_launch(void* const* d_in, const int* in_sizes, int n_in,
                              void* d_out, int out_size, void* d_ws, size_t ws_size,
                              hipStream_t stream) {
    const float* pred   = (const float*)d_in[0];
    const int*   target = (const int*)d_in[1];
    float* partial = (float*)d_ws;        // 256 floats, fully overwritten
    float* out     = (float*)d_out;       // 1 float

    bpr_row_kernel<<<BATCH, THREADS, 0, stream>>>(pred, target, partial);
    bpr_final_kernel<<<1, 32, 0, stream>>>(partial, out);
}